// QuaildGraphCutLoss_32212254720421
// MI455X (gfx1250) — compile-verified
//
#include <hip/hip_runtime.h>
#include <hip/hip_bf16.h>

typedef __attribute__((ext_vector_type(2))) float v2f;
typedef __attribute__((ext_vector_type(4))) float v4f;
typedef __attribute__((ext_vector_type(8))) float v8f;

#define BLK 256
#define DCOLS 256

// Pass 1: streaming column sums of a and b. Each thread owns 4 fixed columns
// (grid stride is a multiple of the 256-wide row), accumulates in fp32 VGPRs,
// then a 4KB LDS reduce produces 512 per-block column partials (256 for a,
// 256 for b) written to ws[blockIdx*512 ..].
__global__ __launch_bounds__(BLK) void colsum_kernel(const float* __restrict__ a,
                                                     const float* __restrict__ b,
                                                     float* __restrict__ ws,
                                                     long long e4) {
    const v4f* __restrict__ A4 = (const v4f*)a;
    const v4f* __restrict__ B4 = (const v4f*)b;

    long long tid = (long long)blockIdx.x * BLK + threadIdx.x;
    long long stride = (long long)gridDim.x * BLK;   // multiple of 64 float4s = 256 floats

    v4f accA = {0.f, 0.f, 0.f, 0.f};
    v4f accB = {0.f, 0.f, 0.f, 0.f};

    #pragma unroll 4
    for (long long i = tid; i < e4; i += stride) {
        v4f va = __builtin_nontemporal_load(&A4[i]);
        v4f vb = __builtin_nontemporal_load(&B4[i]);
        accA += va;
        accB += vb;
    }

    // thread t handles columns 4u..4u+3 where u = t & 63; group g = t >> 6 (0..3)
    __shared__ float red[4 * DCOLS];
    v4f* red4 = (v4f*)red;
    int u = threadIdx.x & 63;
    int g = threadIdx.x >> 6;
    int c = threadIdx.x;

    // reduce A partials across the 4 groups
    red4[g * 64 + u] = accA;
    __syncthreads();
    float blockA = red[c] + red[DCOLS + c] + red[2 * DCOLS + c] + red[3 * DCOLS + c];
    ws[(size_t)blockIdx.x * (2 * DCOLS) + c] = blockA;
    __syncthreads();

    // reduce B partials
    red4[g * 64 + u] = accB;
    __syncthreads();
    float blockB = red[c] + red[DCOLS + c] + red[2 * DCOLS + c] + red[3 * DCOLS + c];
    ws[(size_t)blockIdx.x * (2 * DCOLS) + DCOLS + c] = blockB;
}

// Pass 2: single block. The block-partial sum loop is split 4 ways across
// thread groups (g = tid>>6) with b128 loads: thread (g,u) accumulates v4f
// partials of columns 4u..4u+3 over blocks blk = g, g+4, g+8, ...
// An LDS reduce across g yields per-column totals A_c, B_c; p_c = A_c*B_c is
// then reduced per-wave with one V_WMMA_F32_16X16X4_F32
// (A = {p,0}, B = ones -> D[m,n] = p_m + p_{m+16}), 7 fp32 adds and two lane
// broadcasts; 8 wave sums -> LDS -> scalar.
__global__ __launch_bounds__(BLK) void dot_reduce_kernel(const float* __restrict__ ws,
                                                         float* __restrict__ out,
                                                         int nblocks, float invN) {
    int u = threadIdx.x & 63;
    int g = threadIdx.x >> 6;
    int c = threadIdx.x;

    v4f sa4 = {0.f, 0.f, 0.f, 0.f};
    v4f sb4 = {0.f, 0.f, 0.f, 0.f};
    for (int blk = g; blk < nblocks; blk += 4) {
        const v4f* rowA = (const v4f*)(ws + (size_t)blk * (2 * DCOLS));
        const v4f* rowB = (const v4f*)(ws + (size_t)blk * (2 * DCOLS) + DCOLS);
        sa4 += rowA[u];
        sb4 += rowB[u];
    }

    __shared__ float red[4 * DCOLS];
    v4f* red4 = (v4f*)red;

    red4[g * 64 + u] = sa4;
    __syncthreads();
    float sa = red[c] + red[DCOLS + c] + red[2 * DCOLS + c] + red[3 * DCOLS + c];
    __syncthreads();
    red4[g * 64 + u] = sb4;
    __syncthreads();
    float sb = red[c] + red[DCOLS + c] + red[2 * DCOLS + c] + red[3 * DCOLS + c];

    float p = sa * sb;

    // Wave32 reduction via fp32 WMMA: exact (RNE fp32 accumulate).
    v2f Am; Am.x = p; Am.y = 0.0f;          // A[m,0]=p_m, A[m,2]=p_{m+16}, K=1,3 zero
    v2f Bm; Bm.x = 1.0f; Bm.y = 1.0f;       // all-ones B: layout invariant
    v8f Cz = {};
    v8f Dm = __builtin_amdgcn_wmma_f32_16x16x4_f32(
        /*neg_a=*/false, Am, /*neg_b=*/false, Bm,
        /*c_mod=*/(short)0, Cz, /*reuse_a=*/false, /*reuse_b=*/false);

    // lanes 0-15: sum_{m=0..7}(p_m+p_{m+16}); lanes 16-31: sum_{m=8..15}(...)
    float s = Dm[0] + Dm[1] + Dm[2] + Dm[3] + Dm[4] + Dm[5] + Dm[6] + Dm[7];
    float wsum = __shfl(s, 0, 32) + __shfl(s, 16, 32);   // full 32-lane sum

    __shared__ float wpart[8];
    if ((threadIdx.x & 31) == 0) wpart[threadIdx.x >> 5] = wsum;
    __syncthreads();
    if (threadIdx.x == 0) {
        float tot = 0.f;
        #pragma unroll
        for (int w = 0; w < 8; ++w) tot += wpart[w];
        // loss = mean(row_sums) - (2*lambd*-1) + eps = tot/N + 1
        out[0] = tot * invN + 1.0f;
    }
}

extern "C" void kernel_launch(void* const* d_in, const int* in_sizes, int n_in,
                              void* d_out, int out_size, void* d_ws, size_t ws_size,
                              hipStream_t stream) {
    const float* a = (const float*)d_in[0];
    const float* b = (const float*)d_in[1];
    float* out = (float*)d_out;
    float* ws = (float*)d_ws;

    long long elems = (long long)in_sizes[0];        // 100000 * 256
    long long e4 = elems / 4;
    long long nrows = elems / DCOLS;
    float invN = 1.0f / (float)nrows;

    // Cap block count by available workspace (512 floats per block), deterministic.
    size_t per_block = (size_t)(2 * DCOLS) * sizeof(float);
    int nblocks = 512;
    if (ws_size < (size_t)nblocks * per_block) {
        nblocks = (int)(ws_size / per_block);
        if (nblocks < 1) nblocks = 1;
    }

    colsum_kernel<<<nblocks, BLK, 0, stream>>>(a, b, ws, e4);
    dot_reduce_kernel<<<1, BLK, 0, stream>>>(ws, out, nblocks, invN);
}